// CustomDynamicSimpleVFE_18588618457298
// MI455X (gfx1250) — compile-verified
//
#include <hip/hip_runtime.h>
#include <stdint.h>

#ifndef __has_builtin
#define __has_builtin(x) 0
#endif

// Problem constants (from the reference module)
#define VX 352
#define VY 400
#define VZ 1
#define NBATCH 8

#define SCAN_THREADS 256
#define EPT 8
#define VOX_PER_BLOCK (SCAN_THREADS * EPT)

// Double-buffered async point streaming
#define TILE 256
#define TPB 8                      // tiles per block in pass 1
#define PTS_PER_BLOCK (TILE * TPB)

// Pointer types for the gfx1250 async global->LDS copy builtin:
// signature is (v4i32 addrspace(1)*, v4i32 addrspace(3)*, imm offset, imm cpol)
typedef int v4i __attribute__((vector_size(16)));
typedef __attribute__((address_space(1))) v4i GVec4;  // global
typedef __attribute__((address_space(3))) v4i LVec4;  // LDS

// ---------------------------------------------------------------------------
// Pass 0: init voxel state + output padding (0 / -1 / false per reference)
// ---------------------------------------------------------------------------
__global__ __launch_bounds__(256) void vfe_init(
    int* __restrict__ seg_max, int num_seg,
    float4* __restrict__ pts_out, int4* __restrict__ coords_out,
    float* __restrict__ valid_out, int M) {
  int i = blockIdx.x * 256 + threadIdx.x;
  if (i < num_seg) seg_max[i] = -1;
  if (i < M) {
    pts_out[i]    = make_float4(0.f, 0.f, 0.f, 0.f);
    coords_out[i] = make_int4(-1, -1, -1, -1);
    valid_out[i]  = 0.f;
  }
}

// ---------------------------------------------------------------------------
// Pass 1: per-point voxel key + atomicMax(point index) => keep LAST point.
// Double-buffered async global->LDS pipeline: tile i+1 streams in via the
// CDNA5 async copy engine while tile i is quantized/scattered. All lanes
// always issue (OOB lanes clamp the address) so ASYNCcnt bookkeeping is
// exact: wait(asynccnt<=1) after issuing tile i+1 guarantees tile i landed
// (async loads complete in order).
// ---------------------------------------------------------------------------
__global__ __launch_bounds__(256) void vfe_scatter_last(
    const float* __restrict__ pts, int* __restrict__ seg_max, int M, int N) {
  const int tid = threadIdx.x;
  const int blockStart = blockIdx.x * PTS_PER_BLOCK;

#if __has_builtin(__builtin_amdgcn_global_load_async_to_lds_b128)
  __shared__ float tile[2][TILE * 4];

  // prologue: issue tile 0
  {
    int g0 = blockStart + tid;
    if (g0 >= M) g0 = M - 1;  // clamp: every lane issues
    __builtin_amdgcn_global_load_async_to_lds_b128(
        (GVec4*)(pts + (size_t)g0 * 4), (LVec4*)(&tile[0][tid * 4]),
        /*offset=*/0, /*cpol=*/0);
  }

#pragma unroll
  for (int it = 0; it < TPB; ++it) {
    if (it + 1 < TPB) {
      int gn = blockStart + (it + 1) * TILE + tid;
      if (gn >= M) gn = M - 1;
      __builtin_amdgcn_global_load_async_to_lds_b128(
          (GVec4*)(pts + (size_t)gn * 4),
          (LVec4*)(&tile[(it + 1) & 1][tid * 4]), 0, 0);
      __builtin_amdgcn_s_wait_asynccnt(1);  // tile `it` has landed
    } else {
      __builtin_amdgcn_s_wait_asynccnt(0);
    }
    __syncthreads();

    const int gid = blockStart + it * TILE + tid;
    if (gid < M) {
      float4 p = *reinterpret_cast<float4*>(&tile[it & 1][tid * 4]);
      int cx = (int)((p.x -  0.0f) / 0.2f);
      int cy = (int)((p.y + 40.0f) / 0.2f);
      int cz = (int)((p.z +  3.0f) / 4.0f);
      bool kept = (cx >= 0) & (cx < VX) & (cy >= 0) & (cy < VY) &
                  (cz >= 0) & (cz < VZ);
      if (kept) {
        int bs  = gid / N;
        int key = ((bs * VZ + cz) * VY + cy) * VX + cx;
        atomicMax(seg_max + key, gid);  // global_atomic_max_i32, L2-resident
      }
    }
    __syncthreads();  // buffer (it&1) free for reuse next-next iteration
  }
#else
#pragma unroll
  for (int it = 0; it < TPB; ++it) {
    const int gid = blockStart + it * TILE + tid;
    if (gid < M) {
      float4 p = reinterpret_cast<const float4*>(pts)[gid];
      int cx = (int)((p.x -  0.0f) / 0.2f);
      int cy = (int)((p.y + 40.0f) / 0.2f);
      int cz = (int)((p.z +  3.0f) / 4.0f);
      bool kept = (cx >= 0) & (cx < VX) & (cy >= 0) & (cy < VY) &
                  (cz >= 0) & (cz < VZ);
      if (kept) {
        int bs  = gid / N;
        int key = ((bs * VZ + cz) * VY + cy) * VX + cx;
        atomicMax(seg_max + key, gid);
      }
    }
  }
#endif
}

// ---------------------------------------------------------------------------
// Pass 2a: per-block occupancy counts over voxels
// ---------------------------------------------------------------------------
__global__ __launch_bounds__(SCAN_THREADS) void vfe_block_sums(
    const int* __restrict__ seg_max, unsigned* __restrict__ partials,
    int num_seg) {
  __shared__ unsigned s[SCAN_THREADS];
  const int t = threadIdx.x;
  const int base = blockIdx.x * VOX_PER_BLOCK + t * EPT;
  unsigned cnt = 0;
  if (base + EPT <= num_seg) {
    const int4* q = reinterpret_cast<const int4*>(seg_max + base);
    int4 a = q[0], b = q[1];
    cnt = (unsigned)((a.x >= 0) + (a.y >= 0) + (a.z >= 0) + (a.w >= 0) +
                     (b.x >= 0) + (b.y >= 0) + (b.z >= 0) + (b.w >= 0));
  } else {
#pragma unroll
    for (int e = 0; e < EPT; ++e) {
      int v = base + e;
      if (v < num_seg && seg_max[v] >= 0) cnt++;
    }
  }
  s[t] = cnt;
  __syncthreads();
  for (int off = SCAN_THREADS / 2; off > 0; off >>= 1) {
    if (t < off) s[t] += s[t + off];
    __syncthreads();
  }
  if (t == 0) partials[blockIdx.x] = s[0];
}

// ---------------------------------------------------------------------------
// Pass 2b: exclusive scan of block partials (nblk = 550 <= 1024, one block)
// ---------------------------------------------------------------------------
__global__ __launch_bounds__(1024) void vfe_scan_partials(
    unsigned* __restrict__ partials, int nblk) {
  __shared__ unsigned s[1024];
  const int t = threadIdx.x;
  unsigned v = (t < nblk) ? partials[t] : 0u;
  s[t] = v;
  __syncthreads();
  for (int off = 1; off < 1024; off <<= 1) {
    unsigned x = (t >= off) ? s[t - off] : 0u;
    __syncthreads();
    s[t] += x;
    __syncthreads();
  }
  if (t < nblk) partials[t] = s[t] - v;  // inclusive -> exclusive
}

// ---------------------------------------------------------------------------
// Pass 3: recompute local scan, gather winning point per occupied voxel,
// emit [pts | coords(bs,z,y,x) | valid] at the voxel's global rank.
// ---------------------------------------------------------------------------
__global__ __launch_bounds__(SCAN_THREADS) void vfe_emit(
    const int* __restrict__ seg_max, const unsigned* __restrict__ block_off,
    const float* __restrict__ pts,
    float4* __restrict__ pts_out, int4* __restrict__ coords_out,
    float* __restrict__ valid_out, int num_seg) {
  __shared__ unsigned s[SCAN_THREADS];
  const int t = threadIdx.x;
  const int base = blockIdx.x * VOX_PER_BLOCK + t * EPT;

  int last[EPT];
  unsigned cnt = 0;
  if (base + EPT <= num_seg) {
    const int4* q = reinterpret_cast<const int4*>(seg_max + base);
    int4 a = q[0], b = q[1];
    last[0] = a.x; last[1] = a.y; last[2] = a.z; last[3] = a.w;
    last[4] = b.x; last[5] = b.y; last[6] = b.z; last[7] = b.w;
  } else {
#pragma unroll
    for (int e = 0; e < EPT; ++e) {
      int v = base + e;
      last[e] = (v < num_seg) ? seg_max[v] : -1;
    }
  }
#pragma unroll
  for (int e = 0; e < EPT; ++e) {
    cnt += (unsigned)(last[e] >= 0);
    if (last[e] >= 0) __builtin_prefetch(pts + (size_t)last[e] * 4, 0, 0);
  }

  s[t] = cnt;
  __syncthreads();
  for (int off = 1; off < SCAN_THREADS; off <<= 1) {
    unsigned x = (t >= off) ? s[t - off] : 0u;
    __syncthreads();
    s[t] += x;
    __syncthreads();
  }
  unsigned pos = block_off[blockIdx.x] + (t ? s[t - 1] : 0u);

#pragma unroll
  for (int e = 0; e < EPT; ++e) {
    int sm = last[e];
    if (sm >= 0) {
      int v    = base + e;
      int cx   = v % VX;
      int cy   = (v / VX) % VY;
      int rest = v / (VX * VY);
      int cz   = rest % VZ;
      int bs   = rest / VZ;
      float4 p = reinterpret_cast<const float4*>(pts)[sm];
      pts_out[pos]    = p;
      coords_out[pos] = make_int4(bs, cz, cy, cx);
      valid_out[pos]  = 1.0f;
      pos++;
    }
  }
}

// ---------------------------------------------------------------------------
extern "C" void kernel_launch(void* const* d_in, const int* in_sizes, int n_in,
                              void* d_out, int out_size, void* d_ws,
                              size_t ws_size, hipStream_t stream) {
  const float* pts = (const float*)d_in[0];
  const int C = 4;
  const int M = in_sizes[0] / C;  // 4,000,000
  const int B = NBATCH;
  const int N = M / B;
  const int num_seg = B * VZ * VY * VX;  // 1,126,400

  // workspace: voxel winner table + scan partials (~4.5 MB, L2 resident)
  int* seg_max       = (int*)d_ws;
  unsigned* partials = (unsigned*)(seg_max + num_seg);

  // outputs concatenated flat: pts[M,4] f32 | coords[M,4] i32 | valid[M]
  float* pts_out   = (float*)d_out;
  int* coords_out  = (int*)(pts_out + (size_t)M * 4);
  float* valid_out = (float*)(coords_out + (size_t)M * 4);

  const int init_blocks = (M + 255) / 256;  // M > num_seg, covers both
  vfe_init<<<init_blocks, 256, 0, stream>>>(
      seg_max, num_seg, (float4*)pts_out, (int4*)coords_out, valid_out, M);

  const int scatter_blocks = (M + PTS_PER_BLOCK - 1) / PTS_PER_BLOCK;  // 1954
  vfe_scatter_last<<<scatter_blocks, 256, 0, stream>>>(pts, seg_max, M, N);

  const int nblk = (num_seg + VOX_PER_BLOCK - 1) / VOX_PER_BLOCK;  // 550
  vfe_block_sums<<<nblk, SCAN_THREADS, 0, stream>>>(seg_max, partials, num_seg);
  vfe_scan_partials<<<1, 1024, 0, stream>>>(partials, nblk);
  vfe_emit<<<nblk, SCAN_THREADS, 0, stream>>>(
      seg_max, partials, pts, (float4*)pts_out, (int4*)coords_out, valid_out,
      num_seg);
}